// Yolo_77644418777211
// MI455X (gfx1250) — compile-verified
//
#include <hip/hip_runtime.h>
#include <math.h>
#include <stdint.h>

// Problem constants (from the reference)
#define SS 52
#define CC 80
#define PP 85            // 5 + C
#define NN 255           // 3*P
#define BB 32
#define MM 2048
#define CELLS (SS*SS)            // 2704
#define NOOBJ_BLOCKS (BB*3)      // 96 slices: (b, channel c*P)
#define NPART (MM + NOOBJ_BLOCKS)

typedef __attribute__((ext_vector_type(4))) unsigned int u32x4;
typedef __attribute__((ext_vector_type(4))) int          i32x4;
typedef __attribute__((ext_vector_type(8))) int          i32x8;

#define HAVE_TDM __has_builtin(__builtin_amdgcn_tensor_load_to_lds)

__device__ __forceinline__ float sigm(float v) {
    return 1.0f / (1.0f + expf(-v));
}

// ---------------------------------------------------------------- K2: noobj sweep (TDM-staged)
// First in the file so the disasm snippet shows the tensor_load_to_lds emission.
__global__ __launch_bounds__(256) void k_noobj(
    const float* __restrict__ x, const unsigned char* __restrict__ mask,
    double* __restrict__ partials) {
    __shared__ __align__(16) float tile[CELLS];
    __shared__ double red[256];
    const int blk = blockIdx.x;          // 0..95
    const int t = threadIdx.x;
    // slice base: x[b, c*P, :, :] with b = blk/3, c = blk%3
    const float* src = x + (((size_t)(blk / 3) * NN) + (size_t)(blk % 3) * PP) * CELLS;
    const unsigned char* mrow = mask + (size_t)blk * CELLS;

#if HAVE_TDM
    if (t < 32) {  // wave 0 issues the DMA (TDM ignores EXEC; one instruction per wave)
        unsigned long long ga = (unsigned long long)(uintptr_t)src;
        unsigned ldsoff = (unsigned)(uintptr_t)&tile[0];   // low 32 bits = LDS byte offset
        u32x4 g0;
        g0.x = 1u;                                          // count=1, user mode, no gather
        g0.y = ldsoff;                                      // lds_addr @ D#0[63:32]
        g0.z = (unsigned)(ga & 0xFFFFFFFFull);              // global_addr lo
        g0.w = (unsigned)((ga >> 32) & 0x01FFFFFFull)       // global_addr hi (57-bit)
             | (2u << 30);                                  // type=2 ("image")
        i32x8 g1;
        g1[0] = (2 << 16);                                  // data_size = 4 bytes
        g1[1] = (int)((CELLS & 0xFFFF) << 16);              // tensor_dim0[15:0] @ [63:48]
        g1[2] = (int)((CELLS >> 16) & 0xFFFF) | (1 << 16);  // tensor_dim0 hi | tensor_dim1 = 1
        g1[3] = (int)(CELLS << 16);                         // tile_dim0 = 2704 @ [127:112]
        g1[4] = 1;                                          // tile_dim1 = 1, tile_dim2 = 0
        g1[5] = CELLS;                                      // tensor_dim0_stride lo32 = 2704
        g1[6] = (int)((CELLS & 0xFFFF) << 16);              // stride0 hi = 0 | stride1 lo16
        g1[7] = 0;
        i32x4 z4 = {0, 0, 0, 0};
#if defined(__clang_major__) && (__clang_major__ >= 23)
        i32x8 z8 = {0, 0, 0, 0, 0, 0, 0, 0};
        __builtin_amdgcn_tensor_load_to_lds(g0, g1, z4, z4, z8, 0);
#else
        __builtin_amdgcn_tensor_load_to_lds(g0, g1, z4, z4, 0);
#endif
        __builtin_amdgcn_s_wait_tensorcnt(0);
    }
    __syncthreads();
    double acc = 0.0;
    for (int p = t; p < CELLS; p += 256) {
        if (!mrow[p]) {
            float s = sigm(tile[p]);
            acc += 0.5 * (double)(s * s);                   // LAMBDA_NOOBJ = 0.5
        }
    }
#else
    double acc = 0.0;
    for (int p = t; p < CELLS; p += 256) {
        if (!mrow[p]) {
            float s = sigm(src[p]);
            acc += 0.5 * (double)(s * s);
        }
    }
#endif
    red[t] = acc;
    __syncthreads();
    for (int off = 128; off > 0; off >>= 1) {
        if (t < off) red[t] += red[t + off];
        __syncthreads();
    }
    if (t == 0) partials[MM + blk] = red[0];
}

// ---------------------------------------------------------------- K0: zero dedup mask
__global__ __launch_bounds__(256) void k_zero(unsigned int* __restrict__ w, int nwords) {
    for (int i = blockIdx.x * blockDim.x + threadIdx.x; i < nwords;
         i += gridDim.x * blockDim.x)
        w[i] = 0u;
}

// ---------------------------------------------------------------- K1: per-box gather + terms
__global__ __launch_bounds__(128) void k_box(
    const float* __restrict__ x, const float* __restrict__ n_box,
    const int* __restrict__ n_index, const float* __restrict__ anchor,
    unsigned char* __restrict__ mask, double* __restrict__ partials) {
    __shared__ double red[128];
    const int m = blockIdx.x;
    const int t = threadIdx.x;

    const float bxf = n_box[m * 6 + 0];
    const float clsf = n_box[m * 6 + 1];
    const float cx = n_box[m * 6 + 2];
    const float cy = n_box[m * 6 + 3];
    const float bw = n_box[m * 6 + 4];
    const float bh = n_box[m * 6 + 5];
    const int ni = n_index[m];           // 0..2 (S_INDEX == 0)
    const int b = (int)bxf;
    const float divv = 416.0f / 52.0f;   // 8.0 exactly
    const int ix = (int)(cx / divv);
    const int iy = (int)(cy / divv);
    const int base = ni * PP;

    double term = 0.0;
    if (t < PP) {
        const size_t xi = ((((size_t)b * NN) + base + t) * SS + ix) * SS + iy;
        const float s = sigm(x[xi]);
        if (t == 0) {
            float d = s - 1.0f;
            term = 5.0 * (double)(d * d);                    // LAMBDA_COORD * (conf-1)^2
            mask[((size_t)(b * 3 + ni)) * CELLS + ix * SS + iy] = 1;  // dedup mark
        } else if (t == 1) {
            float ax = (cx - (float)ix * divv) / divv;
            float d = s - ax; term = (double)(d * d);
        } else if (t == 2) {
            float ay = (cy - (float)iy * divv) / divv;
            float d = s - ay; term = (double)(d * d);
        } else if (t == 3) {
            float rw = anchor[2 * ni + 0] * expf(4.0f * s - 2.0f);
            float d = rw * (1.0f / 416.0f) - bw * (1.0f / 416.0f);
            term = (double)(d * d);
        } else if (t == 4) {
            float rh = anchor[2 * ni + 1] * expf(4.0f * s - 2.0f);
            float d = rh * (1.0f / 416.0f) - bh * (1.0f / 416.0f);
            term = (double)(d * d);
        } else {
            float hot = ((t - 5) == (int)clsf) ? 1.0f : 0.0f;
            float d = s - hot; term = (double)(d * d);
        }
    }
    red[t] = term;
    __syncthreads();
    for (int off = 64; off > 0; off >>= 1) {
        if (t < off) red[t] += red[t + off];
        __syncthreads();
    }
    if (t == 0) partials[m] = red[0];
}

// ---------------------------------------------------------------- K3: deterministic final reduce
__global__ __launch_bounds__(256) void k_reduce(
    const double* __restrict__ partials, float* __restrict__ out) {
    __shared__ double red[256];
    double acc = 0.0;
    for (int i = threadIdx.x; i < NPART; i += 256) acc += partials[i];
    red[threadIdx.x] = acc;
    __syncthreads();
    for (int off = 128; off > 0; off >>= 1) {
        if (threadIdx.x < off) red[threadIdx.x] += red[threadIdx.x + off];
        __syncthreads();
    }
    if (threadIdx.x == 0) out[0] = (float)red[0];
}

// ---------------------------------------------------------------- host launcher
extern "C" void kernel_launch(void* const* d_in, const int* in_sizes, int n_in,
                              void* d_out, int out_size, void* d_ws, size_t ws_size,
                              hipStream_t stream) {
    const float* x = (const float*)d_in[0];       // (32, 255, 52, 52) f32
    const float* n_box = (const float*)d_in[1];   // (2048, 6) f32
    const int* n_index = (const int*)d_in[2];     // (2048,) i32
    const float* anchor = (const float*)d_in[3];  // (9, 2) f32

    // workspace layout: [ mask: 96*2704 bytes ][ pad to 256 ][ partials: 2144 doubles ]
    unsigned char* mask = (unsigned char*)d_ws;
    const size_t maskBytes = (size_t)NOOBJ_BLOCKS * CELLS;      // 259,584
    const size_t poff = (maskBytes + 255) & ~(size_t)255;
    double* partials = (double*)((char*)d_ws + poff);

    k_zero<<<64, 256, 0, stream>>>((unsigned int*)mask, (int)(maskBytes / 4));
    k_box<<<MM, 128, 0, stream>>>(x, n_box, n_index, anchor, mask, partials);
    k_noobj<<<NOOBJ_BLOCKS, 256, 0, stream>>>(x, mask, partials);
    k_reduce<<<1, 256, 0, stream>>>(partials, (float*)d_out);
    (void)in_sizes; (void)n_in; (void)out_size; (void)ws_size;
}